// ClusterNet_76699525972277
// MI455X (gfx1250) — compile-verified
//
#include <hip/hip_runtime.h>
#include <hip/hip_bf16.h>
#include <math.h>

typedef float v2f __attribute__((ext_vector_type(2)));
typedef float v8f __attribute__((ext_vector_type(8)));

#define B_N 16384
#define K_N 2048
#define D_N 1000
#define EPS_INV 2.0f     // 1/eps, eps = 0.5
#define NITERS 15
#define NCHUNK 64
#define CHUNK_B (B_N / NCHUNK)   // 256

// ---------------- row squared-norms ----------------
__global__ __launch_bounds__(256) void row_norm_kernel(const float* __restrict__ x,
                                                       float* __restrict__ out, int D) {
    __shared__ float red[256];
    const int row = blockIdx.x;
    const float* p = x + (size_t)row * D;
    float s = 0.f;
    for (int d = threadIdx.x; d < D; d += 256) { float v = p[d]; s += v * v; }
    red[threadIdx.x] = s; __syncthreads();
    for (int off = 128; off > 0; off >>= 1) {
        if (threadIdx.x < off) red[threadIdx.x] += red[threadIdx.x + off];
        __syncthreads();
    }
    if (threadIdx.x == 0) out[row] = red[0];
}

// ---------------- fp32 WMMA GEMM -> dists ----------------
// dists[b,k] = sqrt(max(f2[b] + c2[k] - 2 * <F[b,:], C[k,:]>, 0))
__global__ __launch_bounds__(256) void gemm_dists_kernel(
    const float* __restrict__ F, const float* __restrict__ C,
    const float* __restrict__ f2, const float* __restrict__ c2,
    float* __restrict__ dists)
{
    const int lane  = threadIdx.x & 31;
    const int wave  = threadIdx.x >> 5;   // 0..7
    const int waveM = wave & 3;           // 4 waves along M
    const int waveN = wave >> 2;          // 2 waves along N
    const int m0 = blockIdx.x * 128 + waveM * 32;
    const int n0 = blockIdx.y * 64  + waveN * 32;

    const int r    = lane & 15;           // row within 16-row fragment
    const int koff = (lane >> 4) << 1;    // 0 or 2 (K split across lane halves)

    const float* fa0 = F + (size_t)(m0 + r)      * D_N + koff;
    const float* fa1 = F + (size_t)(m0 + 16 + r) * D_N + koff;
    const float* cb0 = C + (size_t)(n0 + r)      * D_N + koff;
    const float* cb1 = C + (size_t)(n0 + 16 + r) * D_N + koff;

    v8f acc00 = {}, acc01 = {}, acc10 = {}, acc11 = {};

#pragma unroll 2
    for (int k = 0; k < D_N; k += 4) {
        v2f a0 = *(const v2f*)(fa0 + k);
        v2f a1 = *(const v2f*)(fa1 + k);
        v2f b0 = *(const v2f*)(cb0 + k);
        v2f b1 = *(const v2f*)(cb1 + k);
        acc00 = __builtin_amdgcn_wmma_f32_16x16x4_f32(false, a0, false, b0, (short)0, acc00, false, false);
        acc01 = __builtin_amdgcn_wmma_f32_16x16x4_f32(false, a0, false, b1, (short)0, acc01, false, false);
        acc10 = __builtin_amdgcn_wmma_f32_16x16x4_f32(false, a1, false, b0, (short)0, acc10, false, false);
        acc11 = __builtin_amdgcn_wmma_f32_16x16x4_f32(false, a1, false, b1, (short)0, acc11, false, false);
    }

    const int half = lane >> 4;   // C/D layout: lanes 16-31 hold M+8
    const int col0 = lane & 15;

#pragma unroll
    for (int t = 0; t < 4; ++t) {
        v8f acc = (t == 0) ? acc00 : (t == 1) ? acc01 : (t == 2) ? acc10 : acc11;
        const int mt = m0 + ((t >> 1) << 4);
        const int nt = n0 + ((t & 1) << 4);
        const int ccol = nt + col0;
        const float cc = c2[ccol];
#pragma unroll
        for (int vv = 0; vv < 8; ++vv) {
            const int rrow = mt + vv + (half << 3);
            float d2 = f2[rrow] + cc - 2.0f * acc[vv];
            dists[(size_t)rrow * K_N + ccol] = sqrtf(fmaxf(d2, 0.0f));
        }
    }
}

// ---------------- small helpers ----------------
__global__ void set_ones_kernel(float* __restrict__ p) {
    p[blockIdx.x * 256 + threadIdx.x] = 1.0f;
}
__global__ void zero_int_kernel(int* __restrict__ p) {
    p[blockIdx.x * 256 + threadIdx.x] = 0;
}

// ---------------- row pass: v[b] = scale / sum_k exp(-2 d) * u[k] ----------------
__global__ __launch_bounds__(256) void row_pass_kernel(
    const float* __restrict__ dists, const float* __restrict__ u,
    float* __restrict__ v, float scale)
{
    __shared__ float red[256];
    const int b = blockIdx.x;
    const float* row = dists + (size_t)b * K_N;
    float s = 0.f;
    for (int k = threadIdx.x; k < K_N; k += 256)
        s += __expf(-EPS_INV * row[k]) * u[k];
    red[threadIdx.x] = s; __syncthreads();
    for (int off = 128; off > 0; off >>= 1) {
        if (threadIdx.x < off) red[threadIdx.x] += red[threadIdx.x + off];
        __syncthreads();
    }
    if (threadIdx.x == 0) v[b] = scale / red[0];
}

// ---------------- column pass partials: partial[c,k] = sum_{b in chunk c} exp(-2 d)*v[b] ----
__global__ __launch_bounds__(256) void col_partial_kernel(
    const float* __restrict__ dists, const float* __restrict__ v,
    float* __restrict__ partial)
{
    const int k  = blockIdx.x * 256 + threadIdx.x;
    const int b0 = blockIdx.y * CHUNK_B;
    const float* p = dists + (size_t)b0 * K_N + k;
    float s = 0.f;
#pragma unroll 4
    for (int i = 0; i < CHUNK_B; ++i)
        s += __expf(-EPS_INV * p[(size_t)i * K_N]) * v[b0 + i];
    partial[(size_t)blockIdx.y * K_N + k] = s;
}

__global__ void finalize_u_kernel(const float* __restrict__ partial,
                                  float* __restrict__ u, float scale) {
    const int k = blockIdx.x * 256 + threadIdx.x;
    float s = 0.f;
    for (int c = 0; c < NCHUNK; ++c) s += partial[(size_t)c * K_N + k];
    u[k] = scale / s;
}

__global__ void finalize_counts_kernel(const float* __restrict__ partial,
                                       const float* __restrict__ u,
                                       float* __restrict__ out) {
    const int k = blockIdx.x * 256 + threadIdx.x;
    float s = 0.f;
    for (int c = 0; c < NCHUNK; ++c) s += partial[(size_t)c * K_N + k];
    out[k] = u[k] * s;     // soft_counts[k] = u[k] * sum_b M(b,k) * v_final[b]
}

// ---------------- per-row argmins, histograms, per-row loss ----------------
__global__ __launch_bounds__(256) void assign_kernel(
    const float* __restrict__ dists, const float* __restrict__ u,
    float* __restrict__ out_assign, float* __restrict__ loss_pb,
    int* __restrict__ cnt_soft, int* __restrict__ cnt_raw)
{
    __shared__ float sval[256]; __shared__ int sidx[256];
    __shared__ float rval[256]; __shared__ int ridx[256];
    const int b = blockIdx.x;
    const float* row = dists + (size_t)b * K_N;

    float bS = INFINITY; int iS = K_N;
    float bR = INFINITY; int iR = K_N;
    for (int k = threadIdx.x; k < K_N; k += 256) {
        const float d = row[k];
        const float soft = __expf(-EPS_INV * d) * u[k];   // v_final[b] scale preserves argmin
        if (soft < bS || (soft == bS && k < iS)) { bS = soft; iS = k; }
        if (d    < bR || (d    == bR && k < iR)) { bR = d;    iR = k; }
    }
    sval[threadIdx.x] = bS; sidx[threadIdx.x] = iS;
    rval[threadIdx.x] = bR; ridx[threadIdx.x] = iR;
    __syncthreads();
    for (int off = 128; off > 0; off >>= 1) {
        if (threadIdx.x < off) {
            const int o = threadIdx.x + off;
            if (sval[o] < sval[threadIdx.x] ||
                (sval[o] == sval[threadIdx.x] && sidx[o] < sidx[threadIdx.x])) {
                sval[threadIdx.x] = sval[o]; sidx[threadIdx.x] = sidx[o];
            }
            if (rval[o] < rval[threadIdx.x] ||
                (rval[o] == rval[threadIdx.x] && ridx[o] < ridx[threadIdx.x])) {
                rval[threadIdx.x] = rval[o]; ridx[threadIdx.x] = ridx[o];
            }
        }
        __syncthreads();
    }
    if (threadIdx.x == 0) {
        const int a = sidx[0];
        out_assign[b] = (float)a;
        loss_pb[b] = row[a];
        atomicAdd(&cnt_soft[a], 1);
        atomicAdd(&cnt_raw[ridx[0]], 1);
    }
}

__global__ __launch_bounds__(256) void loss_reduce_kernel(const float* __restrict__ loss_pb,
                                                          float* __restrict__ out) {
    __shared__ float red[256];
    float s = 0.f;
    for (int i = threadIdx.x; i < B_N; i += 256) s += loss_pb[i];
    red[threadIdx.x] = s; __syncthreads();
    for (int off = 128; off > 0; off >>= 1) {
        if (threadIdx.x < off) red[threadIdx.x] += red[threadIdx.x + off];
        __syncthreads();
    }
    if (threadIdx.x == 0) out[0] = red[0] / (float)B_N;
}

__global__ void counts_to_float_kernel(const int* __restrict__ cnt, float* __restrict__ out) {
    const int i = blockIdx.x * 256 + threadIdx.x;
    out[i] = (float)cnt[i];
}

// ---------------- launcher ----------------
extern "C" void kernel_launch(void* const* d_in, const int* in_sizes, int n_in,
                              void* d_out, int out_size, void* d_ws, size_t ws_size,
                              hipStream_t stream)
{
    const float* F = (const float*)d_in[0];   // [B_N, D_N]
    const float* C = (const float*)d_in[1];   // [K_N, D_N]
    float* out = (float*)d_out;               // [1 + B + K + K + K]

    char* ws = (char*)d_ws;
    size_t off = 0;
    float* dists   = (float*)(ws + off); off += (size_t)B_N * K_N * sizeof(float);
    float* f2      = (float*)(ws + off); off += (size_t)B_N * sizeof(float);
    float* c2      = (float*)(ws + off); off += (size_t)K_N * sizeof(float);
    float* u       = (float*)(ws + off); off += (size_t)K_N * sizeof(float);
    float* v       = (float*)(ws + off); off += (size_t)B_N * sizeof(float);
    float* partial = (float*)(ws + off); off += (size_t)NCHUNK * K_N * sizeof(float);
    float* loss_pb = (float*)(ws + off); off += (size_t)B_N * sizeof(float);
    int*   cnts    = (int*)  (ws + off); off += (size_t)2 * K_N * sizeof(int);

    // squared norms
    row_norm_kernel<<<B_N, 256, 0, stream>>>(F, f2, D_N);
    row_norm_kernel<<<K_N, 256, 0, stream>>>(C, c2, D_N);

    // fp32 WMMA GEMM -> dists [B_N, K_N]
    gemm_dists_kernel<<<dim3(B_N / 128, K_N / 64), 256, 0, stream>>>(F, C, f2, c2, dists);

    // Sinkhorn in factorized form: Q = diag(u) * exp(-2*dists)^T * diag(v)
    set_ones_kernel<<<K_N / 256, 256, 0, stream>>>(u);
    zero_int_kernel<<<(2 * K_N) / 256, 256, 0, stream>>>(cnts);
    // init column-normalization: v0[b] = 1 / sum_k M(b,k)
    row_pass_kernel<<<B_N, 256, 0, stream>>>(dists, u, v, 1.0f);

    for (int it = 0; it < NITERS; ++it) {
        // u[k] = (1/K) / sum_b M(b,k) v[b]
        col_partial_kernel<<<dim3(K_N / 256, NCHUNK), 256, 0, stream>>>(dists, v, partial);
        finalize_u_kernel<<<K_N / 256, 256, 0, stream>>>(partial, u, 1.0f / (float)K_N);
        // v[b] = (1/B) / sum_k M(b,k) u[k]
        row_pass_kernel<<<B_N, 256, 0, stream>>>(dists, u, v, 1.0f / (float)B_N);
    }
    // final column normalization: v_final[b] = 1 / sum_k u[k] M(b,k)
    row_pass_kernel<<<B_N, 256, 0, stream>>>(dists, u, v, 1.0f);

    // soft_counts[k] = u[k] * sum_b M(b,k) v_final[b]  -> out[1+B .. 1+B+K)
    col_partial_kernel<<<dim3(K_N / 256, NCHUNK), 256, 0, stream>>>(dists, v, partial);
    finalize_counts_kernel<<<K_N / 256, 256, 0, stream>>>(partial, u, out + 1 + B_N);

    // assignments (soft & raw argmin), histograms, per-row loss gather
    assign_kernel<<<B_N, 256, 0, stream>>>(dists, u, out + 1, loss_pb, cnts, cnts + K_N);

    // cluster_loss -> out[0]
    loss_reduce_kernel<<<1, 256, 0, stream>>>(loss_pb, out);

    // cluster_counts, raw_counts as floats -> out[1+B+K ..)
    counts_to_float_kernel<<<(2 * K_N) / 256, 256, 0, stream>>>(cnts, out + 1 + B_N + K_N);
}